// KL_Loss_7696581394552
// MI455X (gfx1250) — compile-verified
//
#include <hip/hip_runtime.h>
#include <math.h>

// ---------------------------------------------------------------------------
// Triangular-histogram symmetric-KL (Bayer phases) for MI455X / gfx1250.
//
// Kernel 1 (hist): streaming histogram, async global->LDS double-buffered
//   pipeline (ASYNCcnt), LDS ds_add_f32 privatized histograms, per-block
//   partials written to d_ws.
// Kernel 2 (reduce): merge partials, KL terms, block reduction -> scalar.
//
// d_ws requirement: 1024 blocks * 4096 floats * 4B = 16 MB.
// ---------------------------------------------------------------------------

#define N_BINS        1024
#define THREADS       256
#define ROWS_PER_BLK  32     // rows of the full 1024-wide image per block
#define SEGS          32     // ROWS_PER_BLK * SEGS == 1024 rows
#define PIX_PER_PHASE (512 * 512)   // n in the reference

// ---- gfx1250 async global->LDS (ASYNCcnt-tracked) -------------------------
// Copies 16 bytes per lane from global address base + gbyte_off to LDS byte offset lds_byte_off.
__device__ __forceinline__ void async_copy_b128(unsigned lds_byte_off,
                                                const float* base,
                                                unsigned gbyte_off) {
    asm volatile("global_load_async_to_lds_b128 %0, %1, %2"
                 :
                 : "v"(lds_byte_off), "v"(gbyte_off), "s"(base)
                 : "memory");
}
__device__ __forceinline__ void wait_async_le1() {
    asm volatile("s_wait_asynccnt 1" ::: "memory");
}
__device__ __forceinline__ void wait_async_0() {
    asm volatile("s_wait_asynccnt 0" ::: "memory");
}

// LDS byte offset of a generic pointer into __shared__ (flat addr low 32 bits
// are the LDS offset per the CDNA5 aperture mapping).
__device__ __forceinline__ unsigned lds_offset(const void* p) {
    return (unsigned)(uintptr_t)p;
}

// ---------------------------------------------------------------------------
// Kernel 1: per-block partial triangular histograms.
// blockIdx.x encodes (img[1b], batch[4b], seg[5b]) : blk = img*512+batch*32+seg
// ---------------------------------------------------------------------------
__global__ __launch_bounds__(THREADS)
void hist_kernel(const float* __restrict__ gt,
                 const float* __restrict__ outp,
                 float* __restrict__ ws) {
    __shared__ float  hist[4 * N_BINS];        // 4 Bayer phases, 16 KB
    __shared__ float4 stage[2][THREADS];       // double-buffered tiles, 8 KB

    const int tid   = threadIdx.x;
    const int blk   = blockIdx.x;
    const int seg   = blk & (SEGS - 1);
    const int batch = (blk >> 5) & 15;
    const int img   = blk >> 9;
    const float* src = img ? outp : gt;

    for (int i = tid; i < 4 * N_BINS; i += THREADS) hist[i] = 0.0f;
    __syncthreads();

    const int r0 = seg * ROWS_PER_BLK;
    const unsigned slice_elems = (unsigned)batch * 1024u * 1024u;
    const unsigned lane_elems  = (unsigned)tid * 4u;

    const unsigned lds0 = lds_offset(&stage[0][tid]);
    const unsigned lds1 = lds_offset(&stage[1][tid]);

    // Prime the pipeline with row r0.
    {
        unsigned g0 = (slice_elems + (unsigned)r0 * 1024u + lane_elems) * 4u;
        async_copy_b128(lds0, src, g0);
    }

    for (int r = 0; r < ROWS_PER_BLK; ++r) {
        const int row = r0 + r;
        if (r + 1 < ROWS_PER_BLK) {
            // Issue next row into the other buffer, then wait for current.
            unsigned g = (slice_elems + (unsigned)(row + 1) * 1024u + lane_elems) * 4u;
            async_copy_b128((r & 1) ? lds0 : lds1, src, g);
            wait_async_le1();
        } else {
            wait_async_0();
        }

        const float4 v = stage[r & 1][tid];   // ds_load_b128, own chunk only

        const int prow = (row & 1) << 1;      // row parity selects phase pair
        float* h_even = &hist[prow * N_BINS];        // col parity 0
        float* h_odd  = &hist[(prow + 1) * N_BINS];  // col parity 1

        const float xs[4] = {v.x, v.y, v.z, v.w};
#pragma unroll
        for (int k = 0; k < 4; ++k) {
            // pos = x (LEFT=0, DELTA=1); idx = clip(floor(pos),0,1022)
            float x  = xs[k];
            float fi = fminf(fmaxf(floorf(x), 0.0f), 1022.0f);
            int   idx  = (int)fi;
            float frac = x - fi;              // may be <0 or >1 OOR, as in ref
            float* h = (k & 1) ? h_odd : h_even;
            atomicAdd(&h[idx],     1.0f - frac);   // ds_add_f32
            atomicAdd(&h[idx + 1], frac);          // ds_add_f32
        }
    }

    __syncthreads();
    // Plain stores of this block's partial (no pre-zero / atomics needed).
    float* dst = ws + (size_t)blk * (4 * N_BINS);
    for (int i = tid; i < 4 * N_BINS; i += THREADS) dst[i] = hist[i];
}

// ---------------------------------------------------------------------------
// Kernel 2: merge partials, symmetric KL, scalar output.
// One block, 1024 threads: thread t owns bin t.
// final = 0.125 * sum_{phase,batch,bin} (hg - ho) * (ln hg - ln ho)
// ---------------------------------------------------------------------------
__global__ __launch_bounds__(1024)
void kld_reduce_kernel(const float* __restrict__ ws, float* __restrict__ out) {
    __shared__ float red[1024];
    const int t = threadIdx.x;
    const float inv_n = 1.0f / (float)PIX_PER_PHASE;

    float acc = 0.0f;
    for (int p = 0; p < 4; ++p) {
        for (int b = 0; b < 16; ++b) {
            const float* wgt = ws + ((size_t)(b) * SEGS) * (4 * N_BINS)
                                  + p * N_BINS + t;
            const float* wot = ws + ((size_t)(16 + b) * SEGS) * (4 * N_BINS)
                                  + p * N_BINS + t;
            float cgt = 0.0f, cot = 0.0f;
#pragma unroll 4
            for (int s = 0; s < SEGS; ++s) {
                cgt += wgt[(size_t)s * (4 * N_BINS)];
                cot += wot[(size_t)s * (4 * N_BINS)];
            }
            // h = count/n, with h==0 -> 1/n (count==0 <=> h==0)
            float hg = (cgt != 0.0f) ? cgt * inv_n : inv_n;
            float ho = (cot != 0.0f) ? cot * inv_n : inv_n;
            acc += (hg - ho) * (logf(hg) - logf(ho));
        }
    }

    red[t] = acc;
    __syncthreads();
    for (int s = 512; s > 0; s >>= 1) {
        if (t < s) red[t] += red[t + s];
        __syncthreads();
    }
    if (t == 0) out[0] = 0.125f * red[0];
}

// ---------------------------------------------------------------------------
extern "C" void kernel_launch(void* const* d_in, const int* in_sizes, int n_in,
                              void* d_out, int out_size, void* d_ws, size_t ws_size,
                              hipStream_t stream) {
    const float* bayer_gt  = (const float*)d_in[0];
    const float* bayer_out = (const float*)d_in[1];
    float* ws  = (float*)d_ws;    // needs 2*16*32*4*1024*4B = 16 MB
    float* out = (float*)d_out;   // 1 float

    hist_kernel<<<2 * 16 * SEGS, THREADS, 0, stream>>>(bayer_gt, bayer_out, ws);
    kld_reduce_kernel<<<1, 1024, 0, stream>>>(ws, out);
}